// GATConv_69423851372800
// MI455X (gfx1250) — compile-verified
//
#include <hip/hip_runtime.h>
#include <math.h>

typedef __attribute__((ext_vector_type(2))) float v2f;
typedef __attribute__((ext_vector_type(8))) float v8f;

#define IN_FEATS 256
#define HF 128          // NUM_HEADS * OUT_FEATS
#define NUM_HEADS 4
#define OUT_FEATS 32
#define NEG_SLOPE 0.2f

static constexpr int LDS_STRIDE = 258;   // 256 + 2 pad: avoids 16-way DS bank conflicts

// ---------------------------------------------------------------------------
// Kernel 1: feat_src = feat @ W_fc^T   via V_WMMA_F32_16X16X4_F32
//   block = 256 threads = 8 waves; one block per 16-row strip of nodes.
//   A-tile (16x256) staged in LDS once, shared by all 8 waves.
//   wave w computes columns [16w, 16w+16) -> 64 wmma ops over K=256.
// ---------------------------------------------------------------------------
__global__ __launch_bounds__(256) void gat_gemm_kernel(
    const float* __restrict__ feat, const float* __restrict__ W_fc,
    float* __restrict__ feat_src) {
  __shared__ float Atile[16 * LDS_STRIDE];
  const int tid = threadIdx.x;
  const int m0 = blockIdx.x * 16;

  // cooperative load: 16 rows x 256 cols, each thread moves 16 floats (4x b128)
  {
    const int r = tid >> 4;          // 0..15
    const int c = (tid & 15) << 4;   // 0,16,...,240
    const float4* s4 = reinterpret_cast<const float4*>(
        feat + (size_t)(m0 + r) * IN_FEATS + c);
    float4* d4 = reinterpret_cast<float4*>(&Atile[r * LDS_STRIDE + c]);
    d4[0] = s4[0]; d4[1] = s4[1]; d4[2] = s4[2]; d4[3] = s4[3];
  }
  __syncthreads();

  const int lane  = tid & 31;
  const int wave  = tid >> 5;          // 0..7
  const int n0    = wave << 4;         // column tile base (output feature)
  const int lrow  = lane & 15;
  const int khalf = (lane >> 4) << 1;  // 0 (lanes 0-15) or 2 (lanes 16-31)

  // A: lane holds {A[m][k+khalf], A[m][k+khalf+1]}  (ISA 16x4 f32 layout)
  const float* Alds = &Atile[lrow * LDS_STRIDE + khalf];
  // B[k][n] = W_fc[n][k]; lane holds {B[k+khalf][n0+lrow], B[k+khalf+1][...]}
  const float* Bg = W_fc + (size_t)(n0 + lrow) * IN_FEATS + khalf;

  v8f acc = {};
#pragma unroll 8
  for (int k = 0; k < IN_FEATS; k += 4) {
    v2f a = *reinterpret_cast<const v2f*>(Alds + k);
    v2f b = *reinterpret_cast<const v2f*>(Bg + k);
    acc = __builtin_amdgcn_wmma_f32_16x16x4_f32(
        /*neg_a=*/false, a, /*neg_b=*/false, b,
        /*c_mod=*/(short)0, acc, /*reuse_a=*/false, /*reuse_b=*/false);
  }

  // D layout: VGPR g -> row m = g + 8*(lane>=16), col = n0 + (lane&15)
  const int mhi = (lane >> 4) << 3;    // 0 or 8
  float* outp = feat_src + (size_t)(m0 + mhi) * HF + n0 + lrow;
#pragma unroll
  for (int g = 0; g < 8; ++g) outp[(size_t)g * HF] = acc[g];
}

// ---------------------------------------------------------------------------
// Kernel 2: per-(node,head) attention logits el/er + init e_max/denom/rst
// ---------------------------------------------------------------------------
__global__ void gat_node_attn_kernel(
    const float* __restrict__ feat_src, const float* __restrict__ attn_l,
    const float* __restrict__ attn_r, float* __restrict__ el,
    float* __restrict__ er, float* __restrict__ e_max,
    float* __restrict__ denom, float* __restrict__ rst, int nh_total) {
  const int idx = blockIdx.x * blockDim.x + threadIdx.x;   // node*4 + h
  if (idx >= nh_total) return;
  const int h = idx & 3;
  const float* fs = feat_src + (size_t)idx * OUT_FEATS;    // node*128 + h*32
  const float* al = attn_l + h * OUT_FEATS;
  const float* ar = attn_r + h * OUT_FEATS;
  float sl = 0.f, sr = 0.f;
#pragma unroll
  for (int f = 0; f < OUT_FEATS; ++f) {
    const float v = fs[f];
    sl += v * al[f];
    sr += v * ar[f];
  }
  el[idx] = sl;
  er[idx] = sr;
  e_max[idx] = -INFINITY;
  denom[idx] = 0.f;
  float* r = rst + (size_t)idx * OUT_FEATS;
#pragma unroll
  for (int f = 0; f < OUT_FEATS; ++f) r[f] = 0.f;
}

// float atomic-max via signed/unsigned integer atomics (monotone bit tricks)
__device__ __forceinline__ void atomicMaxFloat(float* addr, float val) {
  if (val >= 0.0f)
    atomicMax(reinterpret_cast<int*>(addr), __float_as_int(val));
  else
    atomicMin(reinterpret_cast<unsigned int*>(addr), __float_as_uint(val));
}

// ---------------------------------------------------------------------------
// Kernel 3: edge logits (+LeakyReLU) and per-dst segment max
// ---------------------------------------------------------------------------
__global__ void gat_edge_logit_kernel(
    const float* __restrict__ el, const float* __restrict__ er,
    const float* __restrict__ e_w, const float* __restrict__ attn_ew,
    const int* __restrict__ src, const int* __restrict__ dst,
    float* __restrict__ e_buf, float* __restrict__ e_max, int eh_total) {
  const int idx = blockIdx.x * blockDim.x + threadIdx.x;   // edge*4 + h
  if (idx >= eh_total) return;
  const int e = idx >> 2;
  const int h = idx & 3;
  const int s = src[e];
  const int d = dst[e];
  const float ew0 = e_w[(size_t)e * 8 + h * 2 + 0];
  const float ew1 = e_w[(size_t)e * 8 + h * 2 + 1];
  float v = el[s * 4 + h] + er[d * 4 + h] +
            ew0 * attn_ew[h * 2 + 0] + ew1 * attn_ew[h * 2 + 1];
  v = (v > 0.f) ? v : NEG_SLOPE * v;
  e_buf[idx] = v;
  atomicMaxFloat(&e_max[d * 4 + h], v);
}

// ---------------------------------------------------------------------------
// Kernel 4: e_exp = exp(e - max[dst]); denom[dst] += e_exp
// ---------------------------------------------------------------------------
__global__ void gat_edge_exp_kernel(
    const int* __restrict__ dst, const float* __restrict__ e_max,
    float* __restrict__ e_buf, float* __restrict__ denom, int eh_total) {
  const int idx = blockIdx.x * blockDim.x + threadIdx.x;
  if (idx >= eh_total) return;
  const int e = idx >> 2;
  const int h = idx & 3;
  const int d = dst[e];
  const float x = __expf(e_buf[idx] - e_max[d * 4 + h]);
  e_buf[idx] = x;
  atomicAdd(&denom[d * 4 + h], x);
}

// ---------------------------------------------------------------------------
// Kernel 5: weighted scatter-sum. One wave32 per edge; each lane owns a
// float4 feature chunk (128 = 32 lanes x 4), head = lane/8.
// ---------------------------------------------------------------------------
__global__ __launch_bounds__(256) void gat_aggregate_kernel(
    const float* __restrict__ feat_src, const float* __restrict__ e_buf,
    const float* __restrict__ denom, const int* __restrict__ src,
    const int* __restrict__ dst, float* __restrict__ rst, long n_edges) {
  const long gid = (long)blockIdx.x * blockDim.x + threadIdx.x;
  const long e = gid >> 5;
  if (e >= n_edges) return;
  const int lane = (int)(gid & 31);
  const int h = lane >> 3;
  const int s = src[e];
  const int d = dst[e];
  const float a = e_buf[e * 4 + h] / denom[(size_t)d * 4 + h];
  const float4 v =
      reinterpret_cast<const float4*>(feat_src)[(size_t)s * 32 + lane];
  float* outp = rst + (size_t)d * HF + lane * 4;
  atomicAdd(outp + 0, v.x * a);
  atomicAdd(outp + 1, v.y * a);
  atomicAdd(outp + 2, v.z * a);
  atomicAdd(outp + 3, v.w * a);
}

// ---------------------------------------------------------------------------
// Kernel 6: ELU in place on the accumulated output
// ---------------------------------------------------------------------------
__global__ void gat_elu_kernel(float* __restrict__ rst, long total) {
  const long idx = (long)blockIdx.x * blockDim.x + threadIdx.x;
  if (idx >= total) return;
  const float v = rst[idx];
  rst[idx] = (v > 0.f) ? v : (__expf(v) - 1.f);
}

extern "C" void kernel_launch(void* const* d_in, const int* in_sizes, int n_in,
                              void* d_out, int out_size, void* d_ws,
                              size_t ws_size, hipStream_t stream) {
  const float* feat    = (const float*)d_in[0];
  const float* e_w     = (const float*)d_in[1];
  const int*   src     = (const int*)d_in[2];
  const int*   dst     = (const int*)d_in[3];
  const float* W_fc    = (const float*)d_in[4];
  const float* attn_l  = (const float*)d_in[5];
  const float* attn_r  = (const float*)d_in[6];
  const float* attn_ew = (const float*)d_in[7];
  float* out = (float*)d_out;

  const int  N = in_sizes[0] / IN_FEATS;   // 100000 (divisible by 16)
  const long E = in_sizes[2];              // 1600000

  // workspace layout (floats): feat_src | el | er | e_max | denom | e_buf
  float* ws       = (float*)d_ws;
  float* feat_src = ws;                                  // N*128
  float* el       = feat_src + (size_t)N * HF;           // N*4
  float* er       = el + (size_t)N * NUM_HEADS;          // N*4
  float* e_max    = er + (size_t)N * NUM_HEADS;          // N*4
  float* denom    = e_max + (size_t)N * NUM_HEADS;       // N*4
  float* e_buf    = denom + (size_t)N * NUM_HEADS;       // E*4

  // 1) WMMA projection
  gat_gemm_kernel<<<N / 16, 256, 0, stream>>>(feat, W_fc, feat_src);

  // 2) node logits + init (also zeros d_out)
  const int nh = N * NUM_HEADS;
  gat_node_attn_kernel<<<(nh + 255) / 256, 256, 0, stream>>>(
      feat_src, attn_l, attn_r, el, er, e_max, denom, out, nh);

  // 3) edge logits + segment max
  const long eh = E * NUM_HEADS;
  gat_edge_logit_kernel<<<(int)((eh + 255) / 256), 256, 0, stream>>>(
      el, er, e_w, attn_ew, src, dst, e_buf, e_max, (int)eh);

  // 4) exp + segment sum
  gat_edge_exp_kernel<<<(int)((eh + 255) / 256), 256, 0, stream>>>(
      dst, e_max, e_buf, denom, (int)eh);

  // 5) weighted scatter aggregation (one wave per edge)
  const long et = E * 32;
  gat_aggregate_kernel<<<(int)((et + 255) / 256), 256, 0, stream>>>(
      feat_src, e_buf, denom, src, dst, out, E);

  // 6) ELU
  const long nf = (long)N * HF;
  gat_elu_kernel<<<(int)((nf + 255) / 256), 256, 0, stream>>>(out, nf);
}